// VisibilityField_85736137162820
// MI455X (gfx1250) — compile-verified
//
#include <hip/hip_runtime.h>
#include <hip/hip_bf16.h>

typedef __attribute__((ext_vector_type(2))) float v2f;
typedef __attribute__((ext_vector_type(8))) float v8f;

// ---------------------------------------------------------------------------
// Prep kernel: fold intrinsics K and the (1,-1,-1,1) flip into each camera's
// world->camera matrix, producing per-camera records of 16 floats:
//   [0..3]  r0 = fx*m_row0 + cx*r2      (r0 . p == x_pix * z)
//   [4..7]  r1 = -fy*m_row1 + cy*r2     (r1 . p == y_pix * z)
//   [8..11] r2 = -m_row2                (r2 . p == z, flipped)
//   [12]    w  (float)   [13] h (float) [14..15] pad
// Cameras in [B, Bpad) are zero-filled -> z == 0 -> mask always false.
// ---------------------------------------------------------------------------
__global__ void vis_prep_kernel(const float* __restrict__ w2chs,
                                const float* __restrict__ Kin,
                                const int* __restrict__ iw,
                                const int* __restrict__ ih,
                                float* __restrict__ rec,
                                int B, int Bpad) {
    int b = blockIdx.x * blockDim.x + threadIdx.x;
    if (b >= Bpad) return;
    float out[16];
    if (b < B) {
        const float* m = w2chs + b * 16;
        float fx = Kin[b * 9 + 0];
        float cx = Kin[b * 9 + 2];
        float fy = Kin[b * 9 + 4];
        float cy = Kin[b * 9 + 5];
#pragma unroll
        for (int j = 0; j < 4; ++j) {
            float r2 = -m[8 + j];
            out[8 + j] = r2;
            out[0 + j] = fx * m[0 + j] + cx * r2;
            out[4 + j] = -fy * m[4 + j] + cy * r2;
        }
        out[12] = (float)iw[b];
        out[13] = (float)ih[b];
        out[14] = 0.0f;
        out[15] = 0.0f;
    } else {
#pragma unroll
        for (int j = 0; j < 16; ++j) out[j] = 0.0f;
    }
#pragma unroll
    for (int j = 0; j < 16; ++j) rec[b * 16 + j] = out[j];
}

// ---------------------------------------------------------------------------
// Main kernel: one wave handles 16 points; each WMMA_F32_16X16X4_F32 projects
// those 16 points through 4 cameras (12 of 16 A-rows used; rows 6,7,14,15 = 0).
//
// A layout (16x4 f32, 2 VGPRs): lane m<16 holds A[m][K=0,1]; lane m+16 holds
// A[m][K=2,3]. Row packing: M=3c..3c+2 are camera c's (r0,r1,r2) for c=0,1;
// M=8+3c'.. for cameras 2,3.
// B layout (4x16 f32, 2 VGPRs): lane n<16 holds B[K=0,1][n] = (px,py);
// lane n+16 holds B[K=2,3][n] = (pz,1).
// D layout (16x16 f32, 8 VGPRs): lane n<16 VGPR j = D[M=j][n]; lane n+16
// VGPR j = D[M=j+8][n].  => lanes<16 own cameras {c0,c0+1}, lanes>=16 own
// {c0+2,c0+3}, each with (xz,yz,z) in d[0..2] / d[3..5].
//
// Branch-free inner loop: the 4 dead A-rows read a zero slot in LDS with a
// zero stride, so no exec masking occurs anywhere in the loop (WMMA requires
// EXEC == all ones). Partial counts for point n live in lanes n and n+16 and
// are combined with a xor-16 shuffle at the end.
// ---------------------------------------------------------------------------
extern __shared__ float camLds[];

__global__ void vis_main_kernel(const float* __restrict__ pos,
                                const float* __restrict__ rec,
                                float* __restrict__ outp,
                                int P, int Bpad) {
    const int tid = threadIdx.x;
    const int lane = tid & 31;
    const int n = lane & 15;
    const bool hi = lane >= 16;

    // This lane's point (lanes n and n+16 load the same point).
    const int wave = blockIdx.x * (blockDim.x >> 5) + (tid >> 5);
    const int p = wave * 16 + n;
    __builtin_prefetch(pos + (size_t)p * 3, 0, 0);  // global_prefetch_b8

    // Stage camera records into LDS once per block; slot [Bpad*16 .. +16) is
    // an always-zero pad used by the dead A rows.
    const int nfl = Bpad * 16;
    for (int i = tid; i < nfl; i += blockDim.x) camLds[i] = rec[i];
    for (int i = tid; i < 16; i += blockDim.x) camLds[nfl + i] = 0.0f;
    __syncthreads();

    float px = 0.0f, py = 0.0f, pz = 0.0f;
    if (p < P) {
        px = pos[p * 3 + 0];
        py = pos[p * 3 + 1];
        pz = pos[p * 3 + 2];
    }
    v2f bmat;
    bmat.x = hi ? pz : px;
    bmat.y = hi ? 1.0f : py;

    // Lane-constant A-row mapping -> branch-free strided LDS offsets.
    int camq = -1, kind = 0;
    if (n < 6) {
        camq = n / 3;
        kind = n % 3;
    } else if (n >= 8 && n < 14) {
        camq = 2 + (n - 8) / 3;
        kind = (n - 8) % 3;
    }
    const int ksel = hi ? 2 : 0;
    int aoff, astride;
    if (camq >= 0) {
        aoff = camq * 16 + kind * 4 + ksel;  // floats
        astride = 4 * 16;                    // 4 cameras per iteration
    } else {
        aoff = nfl;                          // zero slot
        astride = 0;
    }
    int whoff = (hi ? 2 : 0) * 16 + 12;      // this lane's first camera w,h

    int cnt = 0;
    for (int c0 = 0; c0 < Bpad; c0 += 4) {
        v2f a;
        a.x = camLds[aoff + 0];
        a.y = camLds[aoff + 1];
        aoff += astride;

        v8f c = {};
        v8f d = __builtin_amdgcn_wmma_f32_16x16x4_f32(
            /*neg_a=*/false, a, /*neg_b=*/false, bmat,
            /*c_mod=*/(short)0, c, /*reuse_a=*/false, /*reuse_b=*/false);

#pragma unroll
        for (int q = 0; q < 2; ++q) {
            float xz = d[q * 3 + 0];
            float yz = d[q * 3 + 1];
            float z  = d[q * 3 + 2];
            float w = camLds[whoff + q * 16 + 0];
            float h = camLds[whoff + q * 16 + 1];
            // z>0 required, so bounds checks are division-free.
            bool m = (z > 0.0f) & (xz >= 0.0f) & (xz < w * z) &
                     (yz >= 0.0f) & (yz < h * z);
            cnt += m ? 1 : 0;
        }
        whoff += 4 * 16;
    }

    // Combine the two half-wave partial counts for point n.
    cnt += __shfl_xor(cnt, 16, 32);
    if (!hi && p < P) outp[p] = (float)cnt;
}

extern "C" void kernel_launch(void* const* d_in, const int* in_sizes, int n_in,
                              void* d_out, int out_size, void* d_ws, size_t ws_size,
                              hipStream_t stream) {
    (void)n_in; (void)out_size; (void)ws_size;
    const float* positions = (const float*)d_in[0];  // [N,S,3]
    const float* w2chs     = (const float*)d_in[1];  // [B,4,4]
    const float* K         = (const float*)d_in[2];  // [B,3,3]
    const int*   iw        = (const int*)d_in[3];    // [B]
    const int*   ih        = (const int*)d_in[4];    // [B]

    const int B = in_sizes[3];
    const int P = in_sizes[0] / 3;          // N*S points
    const int Bpad = (B + 3) & ~3;          // multiple of 4 cameras per WMMA

    float* rec = (float*)d_ws;              // Bpad * 16 floats

    // Prep: fold K + flips into camera records.
    {
        dim3 blk(128);
        dim3 grd((Bpad + 127) / 128);
        hipLaunchKernelGGL(vis_prep_kernel, grd, blk, 0, stream,
                           w2chs, K, iw, ih, rec, B, Bpad);
    }

    // Main: 256 threads = 8 waves = 128 points per block.
    {
        const int threads = 256;
        const int pointsPerBlock = (threads / 32) * 16;
        const int grid = (P + pointsPerBlock - 1) / pointsPerBlock;
        const size_t ldsBytes = (size_t)(Bpad * 16 + 16) * sizeof(float);
        hipLaunchKernelGGL(vis_main_kernel, dim3(grid), dim3(threads), ldsBytes,
                           stream, positions, rec, (float*)d_out, P, Bpad);
    }
}